// LocalSpatialEncoding_70901320122520
// MI455X (gfx1250) — compile-verified
//
#include <hip/hip_runtime.h>

// LocalSpatialEncoding (RandLA-Net LocSE) for MI455X / gfx1250.
// B=4, N=65536, K=16, D=8, OUT=8. Store-bandwidth-bound (~402MB out).
// 10->8 projection via V_WMMA_F32_16X16X4_F32 (3 chained K-chunks);
// index tile staged with GLOBAL_LOAD_ASYNC_TO_LDS_B32; NT output stores.

#define BQ 4
#define NQ 65536
#define KQ 16
#define DQ 8
#define OUTQ 8
#define EPSQ 1e-6f
#define SLOPEQ 0.2f

typedef float v2f __attribute__((ext_vector_type(2)));
typedef float v8f __attribute__((ext_vector_type(8)));
typedef float f4  __attribute__((ext_vector_type(4)));

// ---- gfx1250 async copy: global -> LDS, per-lane 4B, ASYNCcnt-tracked ---
__device__ __forceinline__ void async_copy_b32(unsigned int lds_byte_addr,
                                               const void* gptr) {
  asm volatile("global_load_async_to_lds_b32 %0, %1, off"
               :: "v"(lds_byte_addr), "v"((unsigned long long)(size_t)gptr)
               : "memory");
}
__device__ __forceinline__ void wait_async0() {
  asm volatile("s_wait_asynccnt 0x0" ::: "memory");
}

// ---- per-lane B-matrix (10x16 weights, cols 8..15 zero) -----------------
// B layout (f32 4x16, 2 VGPRs): lanes 0-15 hold K=0(.x),K=1(.y); lanes 16-31 K=2,K=3.
struct BMats { v2f b0, b1, b2; };

__device__ __forceinline__ BMats make_b(const float* __restrict__ conv_w, int col, bool hi) {
  const float* wp = conv_w + (col & 7) * 10;
  bool act = (col < OUTQ);
  float w0 = act ? wp[0] : 0.f, w1 = act ? wp[1] : 0.f, w2 = act ? wp[2] : 0.f;
  float w3 = act ? wp[3] : 0.f, w4 = act ? wp[4] : 0.f, w5 = act ? wp[5] : 0.f;
  float w6 = act ? wp[6] : 0.f, w7 = act ? wp[7] : 0.f, w8 = act ? wp[8] : 0.f;
  float w9 = act ? wp[9] : 0.f;
  BMats b;
  b.b0.x = hi ? w2 : w0;  b.b0.y = hi ? w3 : w1;
  b.b1.x = hi ? w6 : w4;  b.b1.y = hi ? w7 : w5;
  b.b2.x = hi ? 0.f : w8; b.b2.y = hi ? 0.f : w9;
  return b;
}

// ---- rf (10-ch relative feature) + 3-chunk WMMA chain -------------------
// rf = [dist, rel(3), center(3), neighbor(3)], padded to K=12.
__device__ __forceinline__ v8f locse_wmma(const float* __restrict__ coords,
                                          int b, int n, int nb, bool hi,
                                          BMats bm, float cval) {
  const float* cc = coords + ((size_t)b * NQ + n) * 3;
  const float* nc = coords + ((size_t)b * NQ + nb) * 3;
  float cx = cc[0], cy = cc[1], cz = cc[2];
  float nx = nc[0], ny = nc[1], nz = nc[2];
  float rx = cx - nx, ry = cy - ny, rz = cz - nz;
  float dd = __builtin_amdgcn_sqrtf(rx * rx + ry * ry + rz * rz);  // raw v_sqrt_f32
  // A layout (f32 16x4, 2 VGPRs): lanes 0-15 K=0(.x),1(.y); lanes 16-31 K=2,3.
  v2f a0, a1, a2;
  a0.x = hi ? ry : dd;  a0.y = hi ? rz : rx;   // K 0..3 : dist, rel
  a1.x = hi ? cz : cx;  a1.y = hi ? nx : cy;   // K 4..7 : center, nb.x
  a2.x = hi ? 0.f : ny; a2.y = hi ? 0.f : nz;  // K 8..11: nb.y, nb.z, pad
  v8f c = {cval, cval, cval, cval, cval, cval, cval, cval};
  c = __builtin_amdgcn_wmma_f32_16x16x4_f32(false, a0, false, bm.b0, (short)0, c, false, false);
  c = __builtin_amdgcn_wmma_f32_16x16x4_f32(false, a1, false, bm.b1, (short)0, c, false, false);
  c = __builtin_amdgcn_wmma_f32_16x16x4_f32(false, a2, false, bm.b2, (short)0, c, false, false);
  return c;
}

// ---- pass 1: per-channel sum / sumsq, deterministic block partials ------
__global__ __launch_bounds__(256) void k_stats(const float* __restrict__ coords,
                                               const int* __restrict__ nbi,
                                               const float* __restrict__ conv_w,
                                               const float* __restrict__ conv_b,
                                               float* __restrict__ partials) {
  int t = threadIdx.x;
  int lane = t & 31;
  int m = lane & 15;             // A row / point-in-tile; also D column (channel)
  bool hi = (lane & 16) != 0;
  BMats bm = make_b(conv_w, m, hi);
  float cval = (m < OUTQ) ? conv_b[m] : 0.f;

  int gw = blockIdx.x * 8 + (t >> 5);   // global wave id, 4096 waves
  int g0 = gw * 64;                     // 64 (b,n) tiles per wave
  float acc_s = 0.f, acc_q = 0.f;
  for (int i = 0; i < 64; ++i) {
    int g = g0 + i;
    int b = g >> 16;
    int n = g & (NQ - 1);
    const int* ip = nbi + ((size_t)g << 4);
    __builtin_prefetch(ip + 64, 0, 0);  // stream-ahead: global_prefetch_b8
    int nb = ip[m];
    v8f dm = locse_wmma(coords, b, n, nb, hi, bm, cval);
#pragma unroll
    for (int r = 0; r < 8; ++r) { float y = dm[r]; acc_s += y; acc_q += y * y; }
  }
  __shared__ float reds[256], redq[256];
  reds[t] = acc_s; redq[t] = acc_q;
  __syncthreads();
#pragma unroll
  for (int off = 128; off >= 16; off >>= 1) {
    if (t < off) { reds[t] += reds[t + off]; redq[t] += redq[t + off]; }
    __syncthreads();
  }
  if (t < 16) {
    partials[blockIdx.x * 32 + t] = reds[t];
    partials[blockIdx.x * 32 + 16 + t] = redq[t];
  }
}

// ---- pass 1b: fold 512 partials -> scale/shift per channel --------------
__global__ void k_finalize(const float* __restrict__ partials,
                           const float* __restrict__ gamma,
                           const float* __restrict__ beta,
                           float* __restrict__ ss) {
  int t = threadIdx.x;
  if (t < 16) {
    float s = 0.f, q = 0.f;
    for (int p = 0; p < 512; ++p) {           // fixed order: deterministic
      s += partials[p * 32 + t];
      q += partials[p * 32 + 16 + t];
    }
    const float invM = 1.0f / (float)((size_t)BQ * NQ * KQ);
    float mean = s * invM;
    float var = q * invM - mean * mean;       // biased variance
    float sc = 0.f, sh = 0.f;
    if (t < OUTQ) {
      sc = gamma[t] * rsqrtf(var + EPSQ);
      sh = beta[t] - mean * sc;
    }
    ss[t] = sc;
    ss[16 + t] = sh;
  }
}

// ---- pass 2: recompute y, BN+LeakyReLU, transpose via LDS, NT stores ----
#define SYSTRIDE 260  // 16*16 + 4 pad: kills bank conflicts, keeps 16B align
__global__ __launch_bounds__(256) void k_main(const float* __restrict__ coords,
                                              const float* __restrict__ features,
                                              const int* __restrict__ nbi,
                                              const float* __restrict__ conv_w,
                                              const float* __restrict__ conv_b,
                                              const float* __restrict__ ss,
                                              float* __restrict__ out1,
                                              float* __restrict__ out2) {
  __shared__ int sidx[256];
  __shared__ __attribute__((aligned(16))) float sy[OUTQ * SYSTRIDE];
  int t = threadIdx.x;
  int tile = blockIdx.x;
  int b = tile >> 12;                 // 4096 tiles per batch
  int n0 = (tile & 4095) << 4;        // 16 consecutive n per block

  // stage the 256-index tile in LDS via gfx1250 async copy (ASYNCcnt)
  async_copy_b32((unsigned int)(size_t)&sidx[t],
                 nbi + (((size_t)b * NQ + n0) << 4) + t);
  wait_async0();
  __syncthreads();

  int lane = t & 31, wave = t >> 5;
  int m = lane & 15;
  bool hi = (lane & 16) != 0;
  BMats bm = make_b(conv_w, m, hi);
  float cval = (m < OUTQ) ? conv_b[m] : 0.f;
  float sc = (m < OUTQ) ? ss[m] : 0.f;
  float sh = (m < OUTQ) ? ss[16 + m] : 0.f;

#pragma unroll
  for (int q = 0; q < 2; ++q) {
    int nl = wave * 2 + q;            // local n: each wave owns 2 of 16
    int n = n0 + nl;
    int nb = sidx[nl * 16 + m];
    v8f dm = locse_wmma(coords, b, n, nb, hi, bm, cval);
    // D layout: lane l, VGPR r -> y[point m=r+8*hi][channel c=l&15]
    if (m < OUTQ) {
      int rowbase = m * SYSTRIDE + nl * 16 + (hi ? 8 : 0);
#pragma unroll
      for (int r = 0; r < 8; ++r) {
        float y = dm[r] * sc + sh;
        y = (y >= 0.f) ? y : SLOPEQ * y;
        sy[rowbase + r] = y;
      }
    }
  }
  __syncthreads();

  const size_t NK = (size_t)NQ * KQ;
  const size_t posb = (size_t)n0 << 4;   // tile offset in (n,k)-flattened row
  // y -> out2[b, c, n, k] and out1[b, 8+c, n, k] : 16B coalesced, NT hint
#pragma unroll
  for (int it = 0; it < 2; ++it) {
    int c = it * 4 + (t >> 6);
    int seg = (t & 63) << 2;
    f4 v = *(const f4*)&sy[c * SYSTRIDE + seg];
    __builtin_nontemporal_store(v, (f4*)&out2[((size_t)(b * OUTQ + c)) * NK + posb + seg]);
    __builtin_nontemporal_store(v, (f4*)&out1[((size_t)(b * 16 + OUTQ + c)) * NK + posb + seg]);
  }
  // neighbor feature gather -> out1[b, d, n, k] : L2-resident gathers, NT 16B stores
#pragma unroll
  for (int it = 0; it < 2; ++it) {
    int flat = it * 256 + t;
    int d = flat >> 6;
    int j = (flat & 63) << 2;
    const float* fp = features + ((size_t)(b * DQ + d)) * NQ;
    f4 v;
    v.x = fp[sidx[j + 0]];
    v.y = fp[sidx[j + 1]];
    v.z = fp[sidx[j + 2]];
    v.w = fp[sidx[j + 3]];
    __builtin_nontemporal_store(v, (f4*)&out1[((size_t)(b * 16 + d)) * NK + posb + j]);
  }
}

extern "C" void kernel_launch(void* const* d_in, const int* in_sizes, int n_in,
                              void* d_out, int out_size, void* d_ws, size_t ws_size,
                              hipStream_t stream) {
  const float* coords   = (const float*)d_in[0];
  const float* features = (const float*)d_in[1];
  const int*   nbi      = (const int*)d_in[2];
  const float* conv_w   = (const float*)d_in[3];
  const float* conv_b   = (const float*)d_in[4];
  const float* gamma    = (const float*)d_in[5];
  const float* beta     = (const float*)d_in[6];

  float* out1 = (float*)d_out;                              // (B,16,N,K)
  float* out2 = out1 + (size_t)BQ * 16 * NQ * KQ;           // (B,8,N,K)

  float* partials = (float*)d_ws;                           // 512*32 floats
  float* ssbuf    = partials + 512 * 32;                    // scale[16], shift[16]

  k_stats   <<<512,            256, 0, stream>>>(coords, nbi, conv_w, conv_b, partials);
  k_finalize<<<1,               32, 0, stream>>>(partials, gamma, beta, ssbuf);
  k_main    <<<BQ * NQ / 16,   256, 0, stream>>>(coords, features, nbi, conv_w, conv_b,
                                                 ssbuf, out1, out2);
}